// DeepGNN_352187319163
// MI455X (gfx1250) — compile-verified
//
#include <hip/hip_runtime.h>

#define NN 100000
#define EE 640000
#define DD 128
#define LL 4

typedef float v2f __attribute__((ext_vector_type(2)));
typedef float v8f __attribute__((ext_vector_type(8)));

__device__ __forceinline__ void atomic_add_f32(float* p, float v) {
    __hip_atomic_fetch_add(p, v, __ATOMIC_RELAXED, __HIP_MEMORY_SCOPE_AGENT);
}

__device__ __forceinline__ v8f wmma_f32x4(v2f a, v2f b, v8f c) {
    // D = A(16x4) x B(4x16) + C, f32 in/out
    return __builtin_amdgcn_wmma_f32_16x16x4_f32(false, a, false, b,
                                                 (short)0, c, false, false);
}

// ---------------- zero fill ----------------
__global__ void zero_kernel(float* __restrict__ p, long n) {
    long i = (long)blockIdx.x * blockDim.x + threadIdx.x;
    if (i < n) p[i] = 0.0f;
}

// ---------------- degree ----------------
__global__ void deg_kernel(const long long* __restrict__ edge_index,
                           float* __restrict__ deg) {
    int e = blockIdx.x * blockDim.x + threadIdx.x;
    if (e < EE) {
        int dst = (int)edge_index[EE + e];
        atomic_add_f32(&deg[dst], 1.0f);
    }
}

// ---------------- input projection: h = x @ Wi^T + bi ----------------
// block = 256 threads = 8 waves; block handles 16 rows; wave w handles cols [16w,16w+16)
__global__ void input_proj_kernel(const float* __restrict__ x,
                                  const float* __restrict__ Wi,
                                  const float* __restrict__ bi,
                                  float* __restrict__ h) {
    const int w    = threadIdx.x >> 5;
    const int lane = threadIdx.x & 31;
    const int r    = lane & 15;
    const int half = lane >> 4;
    const int m0   = blockIdx.x * 16;

    const float* Arow = x  + (size_t)(m0 + r) * DD + 2 * half;
    const float* Brow = Wi + (size_t)(w * 16 + r) * DD + 2 * half; // B[k,n] = Wi[n,k]

    v8f acc = {};
#pragma unroll
    for (int kk = 0; kk < DD; kk += 4) {
        v2f a = *(const v2f*)(Arow + kk);
        v2f b = *(const v2f*)(Brow + kk);
        acc = wmma_f32x4(a, b, acc);
    }

    const int col = w * 16 + r;
    const float bv = bi[col];
#pragma unroll
    for (int v = 0; v < 8; ++v) {
        int row = m0 + v + 8 * half;
        h[(size_t)row * DD + col] = acc[v] + bv;
    }
}

// ---------------- edge scatter: S[dst] += h[src] ----------------
// one wave per edge; lane handles 4 strided floats (coalesced 128B per step)
__global__ void scatter_kernel(const float* __restrict__ h,
                               const long long* __restrict__ edge_index,
                               float* __restrict__ S) {
    const int lane = threadIdx.x & 31;
    const int e    = blockIdx.x * (blockDim.x >> 5) + (threadIdx.x >> 5);
    if (e >= EE) return;
    const int src = (int)edge_index[e];
    const int dst = (int)edge_index[EE + e];
    const float* hs = h + (size_t)src * DD;
    float*       Sd = S + (size_t)dst * DD;
#pragma unroll
    for (int i = 0; i < 4; ++i) {
        float v = hs[i * 32 + lane];
        atomic_add_f32(&Sd[i * 32 + lane], v);
    }
}

// ---------------- fused layer: agg = (inv_deg*S)@Wm^T + bm*(deg>0);
//                  h += relu(agg @ Wu^T + bu) ----------------
__global__ void layer_kernel(const float* __restrict__ S,
                             const float* __restrict__ deg,
                             const float* __restrict__ Wm,
                             const float* __restrict__ bm,
                             const float* __restrict__ Wu,
                             const float* __restrict__ bu,
                             float* __restrict__ h) {
    __shared__ float aggT[16][132]; // padded: stride 132 avoids bank conflicts

    const int w    = threadIdx.x >> 5;
    const int lane = threadIdx.x & 31;
    const int r    = lane & 15;
    const int half = lane >> 4;
    const int m0   = blockIdx.x * 16;
    const int col  = w * 16 + r;

    // ---- GEMM 1: agg tile = (inv_deg ⊙ S) @ Wm^T ----
    {
        const float* Arow = S  + (size_t)(m0 + r) * DD + 2 * half;
        const float* Brow = Wm + (size_t)(w * 16 + r) * DD + 2 * half;
        const float scale = 1.0f / fmaxf(deg[m0 + r], 1.0f);

        v8f acc = {};
#pragma unroll
        for (int kk = 0; kk < DD; kk += 4) {
            v2f a = *(const v2f*)(Arow + kk);
            v2f b = *(const v2f*)(Brow + kk);
            a *= scale;
            acc = wmma_f32x4(a, b, acc);
        }

        const float bmv = bm[col];
#pragma unroll
        for (int v = 0; v < 8; ++v) {
            int row = v + 8 * half;                       // row within tile
            float mask = (deg[m0 + row] > 0.0f) ? 1.0f : 0.0f;
            aggT[row][col] = acc[v] + bmv * mask;
        }
    }

    __syncthreads();

    // ---- GEMM 2: h += relu(agg @ Wu^T + bu) ----
    {
        const float* Brow = Wu + (size_t)(w * 16 + r) * DD + 2 * half;

        v8f acc = {};
#pragma unroll
        for (int kk = 0; kk < DD; kk += 4) {
            v2f a = *(const v2f*)(&aggT[r][kk + 2 * half]);
            v2f b = *(const v2f*)(Brow + kk);
            acc = wmma_f32x4(a, b, acc);
        }

        const float buv = bu[col];
#pragma unroll
        for (int v = 0; v < 8; ++v) {
            int row = m0 + v + 8 * half;
            float val = fmaxf(acc[v] + buv, 0.0f);
            h[(size_t)row * DD + col] += val;
        }
    }
}

extern "C" void kernel_launch(void* const* d_in, const int* in_sizes, int n_in,
                              void* d_out, int out_size, void* d_ws, size_t ws_size,
                              hipStream_t stream) {
    const float*     x    = (const float*)d_in[0];
    const float*     Wi   = (const float*)d_in[1];
    const float*     bi   = (const float*)d_in[2];
    const float*     Wm   = (const float*)d_in[3];
    const float*     bm   = (const float*)d_in[4];
    const float*     Wu   = (const float*)d_in[5];
    const float*     bu   = (const float*)d_in[6];
    const long long* ei   = (const long long*)d_in[7];

    float* h   = (float*)d_out;          // [N, D] result lives in d_out
    float* S   = (float*)d_ws;           // [N, D] scatter accumulator
    float* deg = S + (size_t)NN * DD;    // [N]

    const int mtiles = NN / 16;          // 6250, exact

    // degree (zero then accumulate; ws is not re-zeroed between replays)
    zero_kernel<<<(NN + 255) / 256, 256, 0, stream>>>(deg, NN);
    deg_kernel<<<(EE + 255) / 256, 256, 0, stream>>>(ei, deg);

    // h = x @ Wi^T + bi
    input_proj_kernel<<<mtiles, 256, 0, stream>>>(x, Wi, bi, h);

    const long nd = (long)NN * DD;
    for (int l = 0; l < LL; ++l) {
        zero_kernel<<<(int)((nd + 255) / 256), 256, 0, stream>>>(S, nd);
        scatter_kernel<<<(EE + 7) / 8, 256, 0, stream>>>(h, ei, S);
        layer_kernel<<<mtiles, 256, 0, stream>>>(S, deg,
                                                 Wm + (size_t)l * DD * DD,
                                                 bm + (size_t)l * DD,
                                                 Wu + (size_t)l * DD * DD,
                                                 bu + (size_t)l * DD,
                                                 h);
    }
}